// RelationalTemporalGCN_32100585570778
// MI455X (gfx1250) — compile-verified
//
#include <hip/hip_runtime.h>
#include <math.h>

// ---------------------------------------------------------------------------
// RelationalTemporalGCN on MI455X (gfx1250, wave32).
//
// Cost model: reading x (402 MB) dominates -> ~17us floor at 23.3 TB/s. The
// pool kernel therefore uses b128 non-temporal loads (x is streamed once and
// exceeds the 192 MB L2). All GEMMs total ~5.6 GFLOP on [2048,128]x[128,128]
// shapes -> negligible; run in full fp32 on the matrix pipes via
// V_WMMA_F32_16X16X4_F32 (no precision downgrade needed; compute is not the
// bottleneck, and the fp32 reference stays bit-close).
//
// Graph aggregation uses the fixed stencil from _build_edges (self, past 1..5,
// future 1..5, periodic +-15) in gather form: deterministic (no fp atomics),
// <=13 incoming edges per node, identical semantics to the reference's
// segment_sum / segment_max over dst.
// ---------------------------------------------------------------------------

#define BB 4      // batch
#define DD 2048   // nodes per graph
#define BD 8192   // BB*DD
#define CC 192    // input channels
#define GG 128    // hidden dim (= N_HEADS*H2_DIM)

typedef __attribute__((ext_vector_type(2))) float v2f;
typedef __attribute__((ext_vector_type(4))) float v4f;   // clang vector: valid for NT builtins
typedef __attribute__((ext_vector_type(8))) float v8f;

// ---- Stage 1: adaptive avg pool over 8x8 spatial -> feat[BD, 192] ----------
// One node per block (192 threads). thread = (sq, c4): sq in [0,4) spatial
// quarter, c4 in [0,48) float4 channel group. Each thread streams 16 b128
// non-temporal loads (contiguous 768B per wave iteration), then a 4-way LDS
// reduction folds the spatial quarters.
__global__ void pool_kernel(const v4f* __restrict__ x4, v4f* __restrict__ feat4) {
  const int n  = blockIdx.x;
  const int t  = threadIdx.x;            // 0..191
  const int c4 = t % 48;
  const int sq = t / 48;                 // 0..3
  const v4f* p = x4 + (size_t)n * (64 * 48) + (size_t)sq * (16 * 48) + c4;
  v4f s = {0.f, 0.f, 0.f, 0.f};
#pragma unroll 4
  for (int i = 0; i < 16; ++i) {
    const v4f v = __builtin_nontemporal_load(p + (size_t)i * 48);
    s.x += v.x; s.y += v.y; s.z += v.z; s.w += v.w;
  }
  __shared__ v4f red[4][48];
  red[sq][c4] = s;
  __syncthreads();
  if (sq == 0) {
    const v4f a = red[0][c4], b = red[1][c4], c = red[2][c4], d = red[3][c4];
    v4f o;
    o.x = (a.x + b.x + c.x + d.x) * (1.0f / 64.0f);
    o.y = (a.y + b.y + c.y + d.y) * (1.0f / 64.0f);
    o.z = (a.z + b.z + c.z + d.z) * (1.0f / 64.0f);
    o.w = (a.w + b.w + c.w + d.w) * (1.0f / 64.0f);
    feat4[(size_t)n * 48 + c4] = o;
  }
}

// ---- fp32 WMMA GEMM: C[M,128] = A[M,K] @ W[K,128] (+bias) ------------------
// One wave per 16-row strip: 8 accumulator tiles, A fragment reused across
// the 8 N-tiles. Fragment layouts per CDNA5 ISA 7.12.2.
template <int K>
__global__ void gemm_wmma_f32(const float* __restrict__ A, const float* __restrict__ W,
                              const float* __restrict__ bias, float* __restrict__ Cm,
                              int M) {
  const int wave = (blockIdx.x * blockDim.x + threadIdx.x) >> 5;
  const int lane = threadIdx.x & 31;
  const int m0 = wave << 4;
  if (m0 >= M) return;                       // wave-uniform; EXEC stays all-1s
  const int hs  = lane >> 4;
  const int l16 = lane & 15;
  const int row = m0 + l16;

  v8f acc[8] = {};
  for (int k = 0; k < K; k += 4) {
    const int kk = k + 2 * hs;
    v2f a;
    a.x = A[row * K + kk];
    a.y = A[row * K + kk + 1];
#pragma unroll
    for (int nt = 0; nt < 8; ++nt) {
      const int col = (nt << 4) + l16;
      v2f bf;
      bf.x = W[kk * GG + col];
      bf.y = W[(kk + 1) * GG + col];
      // (neg_a, A, neg_b, B, c_mod, C, reuse_a, reuse_b)
      acc[nt] = __builtin_amdgcn_wmma_f32_16x16x4_f32(
          false, a, false, bf, (short)0, acc[nt], false, false);
    }
  }
#pragma unroll
  for (int nt = 0; nt < 8; ++nt) {
    const int col = (nt << 4) + l16;
    const float bv = bias ? bias[col] : 0.f;
#pragma unroll
    for (int vv = 0; vv < 8; ++vv) {
      const int r = m0 + vv + 8 * hs;
      Cm[(size_t)r * GG + col] = acc[nt][vv] + bv;
    }
  }
}

// ---- LayerNorm over the 128-wide feature axis ------------------------------
__global__ void layernorm128(const float* __restrict__ X, const float* __restrict__ g,
                             const float* __restrict__ b, float* __restrict__ Y) {
  const int row = blockIdx.x;
  const int t = threadIdx.x;                 // 128
  __shared__ float red[GG];
  const float v = X[(size_t)row * GG + t];
  red[t] = v;
  __syncthreads();
  for (int s = 64; s > 0; s >>= 1) { if (t < s) red[t] += red[t + s]; __syncthreads(); }
  const float mu = red[0] * (1.0f / GG);
  __syncthreads();
  const float dv = v - mu;
  red[t] = dv * dv;
  __syncthreads();
  for (int s = 64; s > 0; s >>= 1) { if (t < s) red[t] += red[t + s]; __syncthreads(); }
  const float var = red[0] * (1.0f / GG);
  Y[(size_t)row * GG + t] = dv * rsqrtf(var + 1e-5f) * g[t] + b[t];
}

// ---- RGCN combine: root(+bias) + per-relation mean over incoming edges -----
// Incoming edges of node u (dst-side view of _build_edges):
//   rel0: {u}; rel1: {u+1..u+5}; rel2: {u-1..u-5}; rel3: {u-15,u+15} (clipped)
__global__ void rgcn_combine(const float* __restrict__ root,
                             const float* __restrict__ xr0, const float* __restrict__ xr1,
                             const float* __restrict__ xr2, const float* __restrict__ xr3,
                             float* __restrict__ out) {
  const int n = blockIdx.x;
  const int g = threadIdx.x;                 // 128
  const int t = n % DD;
  const int nb = n - t;
  float acc = root[(size_t)n * GG + g] + xr0[(size_t)n * GG + g];
  float s = 0.f, c = 0.f;
#pragma unroll
  for (int off = 1; off <= 5; ++off)
    if (t + off < DD) { s += xr1[(size_t)(nb + t + off) * GG + g]; c += 1.f; }
  if (c > 0.f) acc += s / c;
  s = 0.f; c = 0.f;
#pragma unroll
  for (int off = 1; off <= 5; ++off)
    if (t - off >= 0) { s += xr2[(size_t)(nb + t - off) * GG + g]; c += 1.f; }
  if (c > 0.f) acc += s / c;
  s = 0.f; c = 0.f;
  if (t >= 15)     { s += xr3[(size_t)(nb + t - 15) * GG + g]; c += 1.f; }
  if (t + 15 < DD) { s += xr3[(size_t)(nb + t + 15) * GG + g]; c += 1.f; }
  if (c > 0.f) acc += s / c;
  out[(size_t)n * GG + g] = acc;
}

// ---- TransformerConv edge attention: one wave32 per (node, head) -----------
// H2_DIM == 32 == wave width; head dot-products are 5-step shfl_xor trees.
// 13 fixed candidate edge slots with validity masks (no dynamic indexing).
__global__ void attn_kernel(const float* __restrict__ q, const float* __restrict__ k,
                            const float* __restrict__ v, const float* __restrict__ skip,
                            float* __restrict__ out2) {
  const int n = blockIdx.x;
  const int gidx = threadIdx.x;              // head*32 + dim
  const int t = n % DD;
  const int nb = n - t;

  int srcs[13]; float valid[13];
  srcs[0] = t; valid[0] = 1.f;
#pragma unroll
  for (int off = 1; off <= 5; ++off) {
    srcs[off]     = (t + off < DD) ? t + off : t; valid[off]     = (t + off < DD) ? 1.f : 0.f;
    srcs[5 + off] = (t - off >= 0) ? t - off : t; valid[5 + off] = (t - off >= 0) ? 1.f : 0.f;
  }
  srcs[11] = (t >= 15)     ? t - 15 : t; valid[11] = (t >= 15)     ? 1.f : 0.f;
  srcs[12] = (t + 15 < DD) ? t + 15 : t; valid[12] = (t + 15 < DD) ? 1.f : 0.f;

  const float qv = q[(size_t)n * GG + gidx];
  float logit[13];
  float mx = -1e30f;
#pragma unroll
  for (int e = 0; e < 13; ++e) {
    float p = qv * k[(size_t)(nb + srcs[e]) * GG + gidx];
#pragma unroll
    for (int s = 16; s > 0; s >>= 1) p += __shfl_xor(p, s, 32);  // head dot
    p *= 0.17677669529663687f;               // 1/sqrt(32)
    p = (valid[e] > 0.f) ? p : -1e30f;       // invalid -> weight exactly 0
    logit[e] = p;
    mx = fmaxf(mx, p);
  }
  float denom = 0.f, acc = 0.f;
#pragma unroll
  for (int e = 0; e < 13; ++e) {
    const float a = __expf(logit[e] - mx);
    denom += a;
    acc += a * v[(size_t)(nb + srcs[e]) * GG + gidx];
  }
  out2[(size_t)n * GG + gidx] = acc / denom + skip[(size_t)n * GG + gidx];
}

// ---- BatchNorm (training stats over the 2048 nodes of each graph) ----------
__global__ void bn_stats(const float* __restrict__ X, float* __restrict__ mu,
                         float* __restrict__ var) {
  const int b = blockIdx.x >> 7;             // blockIdx = b*128 + g
  const int g = blockIdx.x & 127;
  const int tid = threadIdx.x;               // 256
  float s = 0.f, s2 = 0.f;
  for (int t = tid; t < DD; t += 256) {
    const float xv = X[(size_t)(b * DD + t) * GG + g];
    s += xv; s2 += xv * xv;
  }
  __shared__ float rs[256], rs2[256];
  rs[tid] = s; rs2[tid] = s2;
  __syncthreads();
  for (int st = 128; st > 0; st >>= 1) {
    if (tid < st) { rs[tid] += rs[tid + st]; rs2[tid] += rs2[tid + st]; }
    __syncthreads();
  }
  if (tid == 0) {
    const float m = rs[0] * (1.0f / DD);
    mu[blockIdx.x] = m;
    var[blockIdx.x] = rs2[0] * (1.0f / DD) - m * m;
  }
}

__global__ void bn_apply_lrelu(const float* __restrict__ X, const float* __restrict__ mu,
                               const float* __restrict__ var, const float* __restrict__ gam,
                               const float* __restrict__ bet, float* __restrict__ Y) {
  const int n = blockIdx.x;
  const int g = threadIdx.x;                 // 128
  const int b = n / DD;
  const float m = mu[b * GG + g];
  const float y = (X[(size_t)n * GG + g] - m) * rsqrtf(var[b * GG + g] + 1e-5f) * gam[g] + bet[g];
  Y[(size_t)n * GG + g] = (y < 0.f) ? 0.01f * y : y;
}

// ---------------------------------------------------------------------------
extern "C" void kernel_launch(void* const* d_in, const int* in_sizes, int n_in,
                              void* d_out, int out_size, void* d_ws, size_t ws_size,
                              hipStream_t stream) {
  (void)in_sizes; (void)n_in; (void)out_size; (void)ws_size;
  const float* x       = (const float*)d_in[0];
  // d_in[1]/d_in[2] (edge_index/edge_type) are the fixed stencil from
  // _build_edges; gathered implicitly in rgcn_combine / attn_kernel.
  const float* fe_fc_w = (const float*)d_in[3];
  const float* fe_fc_b = (const float*)d_in[4];
  const float* fe_ln_g = (const float*)d_in[5];
  const float* fe_ln_b = (const float*)d_in[6];
  const float* w_rel   = (const float*)d_in[7];
  const float* w_root  = (const float*)d_in[8];
  const float* rgcn_b  = (const float*)d_in[9];
  const float* wq = (const float*)d_in[10];
  const float* bq = (const float*)d_in[11];
  const float* wk = (const float*)d_in[12];
  const float* bk = (const float*)d_in[13];
  const float* wv = (const float*)d_in[14];
  const float* bvv = (const float*)d_in[15];
  const float* wskip = (const float*)d_in[16];
  const float* bskip = (const float*)d_in[17];
  const float* bn_g = (const float*)d_in[18];
  const float* bn_b = (const float*)d_in[19];

  float* ws = (float*)d_ws;
  size_t off = 0;
  float* feat = ws + off; off += (size_t)BD * CC;   // pooled features
  float* nf   = ws + off; off += (size_t)BD * GG;   // layer input
  float* bufA = ws + off; off += (size_t)BD * GG;   // hfe / root-out / out2
  float* nf1  = ws + off; off += (size_t)BD * GG;   // after RGCN
  float* xr[4];
  for (int r = 0; r < 4; ++r) { xr[r] = ws + off; off += (size_t)BD * GG; }
  float* qb = ws + off; off += (size_t)BD * GG;
  float* kb = ws + off; off += (size_t)BD * GG;
  float* vb = ws + off; off += (size_t)BD * GG;
  float* sb = ws + off; off += (size_t)BD * GG;
  float* bmu  = ws + off; off += (size_t)BB * GG;
  float* bvar = ws + off; off += (size_t)BB * GG;

  const dim3 ggrid(64), gblock(256);               // 512 waves -> 8192 rows

  // Feature extractor: pool (bandwidth stage) -> FC(+bias) -> LayerNorm
  pool_kernel<<<BD, CC, 0, stream>>>((const v4f*)x, (v4f*)feat);
  gemm_wmma_f32<CC><<<ggrid, gblock, 0, stream>>>(feat, fe_fc_w, fe_fc_b, bufA, BD);
  layernorm128<<<BD, GG, 0, stream>>>(bufA, fe_ln_g, fe_ln_b, nf);

  for (int l = 0; l < 2; ++l) {
    // RGCN: root projection (+bias) and 4 relation projections
    gemm_wmma_f32<GG><<<ggrid, gblock, 0, stream>>>(nf, w_root + (size_t)l * GG * GG,
                                                    rgcn_b + l * GG, bufA, BD);
    for (int r = 0; r < 4; ++r)
      gemm_wmma_f32<GG><<<ggrid, gblock, 0, stream>>>(
          nf, w_rel + (size_t)(l * 4 + r) * GG * GG, nullptr, xr[r], BD);
    rgcn_combine<<<BD, GG, 0, stream>>>(bufA, xr[0], xr[1], xr[2], xr[3], nf1);

    // TransformerConv projections
    gemm_wmma_f32<GG><<<ggrid, gblock, 0, stream>>>(nf1, wq + (size_t)l * GG * GG, bq + l * GG, qb, BD);
    gemm_wmma_f32<GG><<<ggrid, gblock, 0, stream>>>(nf1, wk + (size_t)l * GG * GG, bk + l * GG, kb, BD);
    gemm_wmma_f32<GG><<<ggrid, gblock, 0, stream>>>(nf1, wv + (size_t)l * GG * GG, bvv + l * GG, vb, BD);
    gemm_wmma_f32<GG><<<ggrid, gblock, 0, stream>>>(nf1, wskip + (size_t)l * GG * GG, bskip + l * GG, sb, BD);
    attn_kernel<<<BD, GG, 0, stream>>>(qb, kb, vb, sb, bufA);

    // BatchNorm (training-mode stats) + leaky ReLU; layer 1 writes d_out.
    bn_stats<<<BB * GG, 256, 0, stream>>>(bufA, bmu, bvar);
    float* dst = (l == 1) ? (float*)d_out : nf;
    bn_apply_lrelu<<<BD, GG, 0, stream>>>(bufA, bmu, bvar, bn_g + l * GG, bn_b + l * GG, dst);
  }
}